// RefLocal3_65670049956459
// MI455X (gfx1250) — compile-verified
//
#include <hip/hip_runtime.h>

typedef __attribute__((ext_vector_type(2))) float v2f;
typedef __attribute__((ext_vector_type(8))) float v8f;

#define BATCH 2
#define HH    256
#define WW    256
#define CC    32
#define BIN   32
#define KW    5
#define PAD   2
#define TILE  16   // pixels per wave (M dimension of WMMA)
#define RW    24   // ref tile width:  cols w0-2 .. w0+21
#define VW    32   // rv  tile width:  cols w0-2 .. w0+29

__device__ __forceinline__ uint32_t lds_of(const void* p) {
  return (uint32_t)(uintptr_t)p;           // generic LDS addr: offset in low 32 bits
}

// CDNA5 async DMA: global -> LDS, 16 bytes per lane, tracked by ASYNCcnt.
__device__ __forceinline__ void ld_async_b128(uint32_t lds_addr, const float* gbase,
                                              int byte_off) {
  asm volatile("global_load_async_to_lds_b128 %0, %1, %2"
               :: "v"(lds_addr), "v"(byte_off), "s"(gbase)
               : "memory");
}

// Copy nbytes (multiple of 16, 16B-aligned) from gptr to LDS, wave-cooperative.
__device__ __forceinline__ void copy_async(uint32_t lds_base, const float* gptr,
                                           int nbytes, int lane) {
  for (int off = lane * 16; off < nbytes; off += 32 * 16)
    ld_async_b128(lds_base + off, gptr, off);
}

__device__ __forceinline__ int imin(int a, int b) { return a < b ? a : b; }
__device__ __forceinline__ int imax(int a, int b) { return a > b ? a : b; }

// One wave32 per workgroup; each wave produces 16 output pixels of one row.
__global__ __launch_bounds__(32) void local_attn_wmma_f32(
    const float* __restrict__ gmain,
    const float* __restrict__ gref,
    const float* __restrict__ grv,
    float* __restrict__ gout)
{
  const int lane  = threadIdx.x;           // 0..31 (wave32)
  const int gid   = blockIdx.x;
  const int strip = gid & 15;              // W / TILE = 16 strips
  const int h     = (gid >> 4) & 255;
  const int b     = gid >> 12;
  const int w0    = strip * TILE;

  __shared__ __align__(16) float sMain[TILE][CC];       //  2 KB
  __shared__ __align__(16) float sRef [KW][RW][CC];     // 15 KB  [di][x][c]
  __shared__ __align__(16) float sRV  [KW][VW][BIN];    // 20 KB  [di][x][bin]
  __shared__ __align__(16) float sA   [TILE][CC];       //  2 KB  banded attn matrix
  __shared__ __align__(16) float sD   [2][TILE][TILE];  //  2 KB  D bounce / out staging

  // ---- zero banded-A staging (only 5-wide band is rewritten later) ----
  {
    float4 z4 = make_float4(0.f, 0.f, 0.f, 0.f);
    for (int idx = lane; idx < TILE * CC / 4; idx += 32) ((float4*)sA)[idx] = z4;
  }

  // ---- border workgroups: pre-zero halo tiles so async copies land on zeros ----
  const bool border = (h < PAD) | (h >= HH - PAD) | (strip == 0) | (strip == 15);
  if (border) {
    float4 z4 = make_float4(0.f, 0.f, 0.f, 0.f);
    for (int idx = lane; idx < KW * RW * CC / 4; idx += 32) ((float4*)sRef)[idx] = z4;
    for (int idx = lane; idx < KW * VW * BIN / 4; idx += 32) ((float4*)sRV)[idx] = z4;
  }
  asm volatile("s_wait_dscnt 0x0" ::: "memory");  // zeros committed before DMA writes

  // ---- stage tiles into LDS via async global->LDS DMA (raw fp32 bytes) ----
  const size_t imgBase = ((size_t)b * HH + h) * WW;  // pixel index of (b,h,0)
  copy_async(lds_of(&sMain[0][0]), gmain + (imgBase + w0) * CC, TILE * CC * 4, lane);

  const int x0 = w0 - PAD;
  const int gx0 = imax(0, x0);
  #pragma unroll
  for (int di = 0; di < KW; ++di) {
    int gr = h + di - PAD;
    if ((unsigned)gr < HH) {
      size_t rowPix = ((size_t)b * HH + gr) * WW + gx0;
      int rx1 = imin(WW, x0 + RW);
      copy_async(lds_of(&sRef[di][gx0 - x0][0]), gref + rowPix * CC,
                 (rx1 - gx0) * CC * 4, lane);
      int vx1 = imin(WW, x0 + VW);
      copy_async(lds_of(&sRV[di][gx0 - x0][0]), grv + rowPix * BIN,
                 (vx1 - gx0) * BIN * 4, lane);
    }
  }
  asm volatile("s_wait_asynccnt 0x0" ::: "memory");
  __syncthreads();

  // Fragment lane mapping per CDNA5 ISA (f32 WMMA 16x16x4):
  //  A 16x4 : m = lane&15 ; VGPR j holds K = 2*grp + j within each K-block of 4
  //  B 4x16 : n = lane&15 ; VGPR j holds K = 2*grp + j
  //  C/D    : n = lane&15 ; VGPR v holds M = v + 8*grp
  const int m   = lane & 15;
  const int n   = lane & 15;
  const int grp = lane >> 4;

  // ---- hoisted A fragments (main pixels), 8 K-chunks of 4 ----
  v2f av[8];
  #pragma unroll
  for (int kt = 0; kt < 8; ++kt)
    av[kt] = *(const v2f*)&sMain[m][4 * kt + 2 * grp];

  // ---------------- stage 1: attention logits (banded QK^T) ----------------
  float logit[KW * KW];
  #pragma unroll
  for (int di = 0; di < KW; ++di) {
    v8f d1 = {}, d2 = {};                  // windows at x=0..15 and x=8..23
    #pragma unroll
    for (int kt = 0; kt < 8; ++kt) {
      int k0 = 4 * kt + 2 * grp;
      v2f b1v = *(const v2f*)&sRef[di][n    ][k0];
      v2f b2v = *(const v2f*)&sRef[di][n + 8][k0];
      d1 = __builtin_amdgcn_wmma_f32_16x16x4_f32(false, av[kt], false, b1v,
                                                 (short)0, d1, false, false);
      d2 = __builtin_amdgcn_wmma_f32_16x16x4_f32(false, av[kt], false, b2v,
                                                 (short)0, d2, false, false);
    }
    #pragma unroll
    for (int v = 0; v < 8; ++v) {
      sD[0][v + 8 * grp][n] = d1[v];
      sD[1][v + 8 * grp][n] = d2[v];
    }
    __syncthreads();
    if (lane < 16) {                       // extract the 5-wide diagonal band
      #pragma unroll
      for (int dj = 0; dj < KW; ++dj) {
        int c = m + dj;                    // tile-col index of the neighbor
        logit[di * KW + dj] = (c < 16) ? sD[0][m][c] : sD[1][m][c - 8];
      }
    }
    __syncthreads();
  }

  // ---------------- softmax over 25 (lanes 0..15 own one pixel each) ----------------
  float wgt[KW * KW];
  if (lane < 16) {
    float mx = logit[0];
    #pragma unroll
    for (int j = 1; j < KW * KW; ++j) mx = fmaxf(mx, logit[j]);
    float s = 0.f;
    #pragma unroll
    for (int j = 0; j < KW * KW; ++j) { wgt[j] = __expf(logit[j] - mx); s += wgt[j]; }
    float inv = 1.f / s;
    #pragma unroll
    for (int j = 0; j < KW * KW; ++j) wgt[j] *= inv;
  }

  // ---------------- stage 2: banded attn x values (K-chunks 0..4 cover k<20) ----------------
  v8f acc0 = {}, acc1 = {};
  #pragma unroll
  for (int di = 0; di < KW; ++di) {
    if (lane < 16) {                       // band positions p..p+4 are di-invariant
      #pragma unroll
      for (int dj = 0; dj < KW; ++dj)
        sA[m][m + dj] = wgt[di * KW + dj];
    }
    __syncthreads();
    #pragma unroll
    for (int kt = 0; kt < 5; ++kt) {       // k0 max = 18, +1 = 19 < 20: full band
      int k0 = 4 * kt + 2 * grp;
      v2f a2v = *(const v2f*)&sA[m][k0];
      v2f bva, bvb;
      bva[0] = sRV[di][k0    ][n];
      bva[1] = sRV[di][k0 + 1][n];
      bvb[0] = sRV[di][k0    ][16 + n];
      bvb[1] = sRV[di][k0 + 1][16 + n];
      acc0 = __builtin_amdgcn_wmma_f32_16x16x4_f32(false, a2v, false, bva,
                                                   (short)0, acc0, false, false);
      acc1 = __builtin_amdgcn_wmma_f32_16x16x4_f32(false, a2v, false, bvb,
                                                   (short)0, acc1, false, false);
    }
    __syncthreads();
  }

  // ---------------- write out: stage in LDS, coalesced b128 stores ----------------
  float* sOut = (float*)sD;                // 16 px * 32 bins f32 == 2 KB
  #pragma unroll
  for (int v = 0; v < 8; ++v) {
    int mm = v + 8 * grp;
    sOut[mm * BIN + n]      = acc0[v];
    sOut[mm * BIN + 16 + n] = acc1[v];
  }
  __syncthreads();
  const size_t outBase = (imgBase + w0) * BIN;
  const float4* s4 = (const float4*)sOut;
  float4* g4 = (float4*)(gout + outBase);
  #pragma unroll
  for (int t = 0; t < 4; ++t)
    g4[t * 32 + lane] = s4[t * 32 + lane];
}

extern "C" void kernel_launch(void* const* d_in, const int* in_sizes, int n_in,
                              void* d_out, int out_size, void* d_ws, size_t ws_size,
                              hipStream_t stream) {
  const float* gmain = (const float*)d_in[0];
  const float* gref  = (const float*)d_in[1];
  const float* grv   = (const float*)d_in[2];
  float* gout = (float*)d_out;
  dim3 grid(BATCH * HH * (WW / TILE));     // 2*256*16 = 8192 single-wave WGs
  local_attn_wmma_f32<<<grid, 32, 0, stream>>>(gmain, gref, grv, gout);
}